// LengthRegulator_37022618092115
// MI455X (gfx1250) — compile-verified
//
#include <hip/hip_runtime.h>

// Problem constants (from reference): B=64, T=512, D=384, MAX_LEN=4096.
#define NB 64
#define NT 512
#define ND 384
#define ML 4096

// ---- CDNA5 async global<->LDS DMA availability --------------------------------
#if __has_builtin(__builtin_amdgcn_global_load_async_to_lds_b128) && \
    __has_builtin(__builtin_amdgcn_global_store_async_from_lds_b128)
#define USE_ASYNC_LDS 1
#else
#define USE_ASYNC_LDS 0
#endif

// b128 async ops are typed as 4 x i32 vectors; global = AS1, LDS = AS3.
typedef int v4i __attribute__((ext_vector_type(4)));
typedef __attribute__((address_space(1))) v4i gv4i;
typedef __attribute__((address_space(3))) v4i lv4i;

static __device__ __forceinline__ void wait_asynccnt0() {
#if __has_builtin(__builtin_amdgcn_s_wait_asynccnt)
  __builtin_amdgcn_s_wait_asynccnt(0);
#else
  asm volatile("s_wait_asynccnt 0" ::: "memory");
#endif
}

// ---- Kernel A: per-batch inclusive scan of durations --------------------------
// grid = NB blocks, 512 threads. Writes csum[b*NT+t], mel[b], and mel as f32
// into the tail of d_out (tuple output #2, concatenated after `out`).
__global__ __launch_bounds__(NT) void lr_scan(const int* __restrict__ dur,
                                              int* __restrict__ csum,
                                              int* __restrict__ mel,
                                              float* __restrict__ melf) {
  __shared__ int s[NT];
  const int b = blockIdx.x;
  const int t = threadIdx.x;
  s[t] = dur[b * NT + t];
  __syncthreads();
#pragma unroll
  for (int off = 1; off < NT; off <<= 1) {
    int v = (t >= off) ? s[t - off] : 0;
    __syncthreads();
    s[t] += v;
    __syncthreads();
  }
  csum[b * NT + t] = s[t];
  if (t == NT - 1) {
    mel[b] = s[t];
    melf[b] = (float)s[t];  // tuple output concatenation promotes int32 -> f32
  }
}

// ---- Kernel B: one wave32 per output row --------------------------------------
// row = b*ML + p. If p >= mel[b]: zero-fill (96 float4 / wave, coalesced).
// Else: branchless binary search (9 steps, wave-uniform, L2-resident csum),
// then DMA the 1536 B row: 3x async b128 global->LDS, wait, 3x b128 LDS->global.
__global__ __launch_bounds__(256) void lr_expand(const float* __restrict__ x,
                                                 const int* __restrict__ csum,
                                                 const int* __restrict__ mel,
                                                 float* __restrict__ out) {
  const int lane = threadIdx.x & 31;
  const int wave = threadIdx.x >> 5;
  const int row  = blockIdx.x * 8 + wave;   // wave-uniform
  const int b    = row >> 12;               // row / 4096
  const int p    = row & (ML - 1);          // row % 4096

  float* dst = out + (size_t)row * ND;
  const int m = mel[b];

#if USE_ASYNC_LDS
  __shared__ float lbuf[8 * ND];            // 12 KB, one 1536 B slot per wave
#endif

  if (p >= m) {                             // wave-uniform branch: masked tail
    const float4 z = make_float4(0.f, 0.f, 0.f, 0.f);
    float4* d4 = (float4*)dst;
    d4[lane]      = z;
    d4[lane + 32] = z;
    d4[lane + 64] = z;
    return;
  }

  // searchsorted(csum, p, side='right') == count of csum[j] <= p, branchless.
  const int* c = csum + b * NT;
  int idx = 0;
#pragma unroll
  for (int s = 256; s >= 1; s >>= 1)
    if (c[idx + s - 1] <= p) idx += s;      // idx <= 511 guaranteed when p < m

  const float* src = x + (size_t)(b * NT + idx) * ND;

#if USE_ASYNC_LDS
  char* lbase = (char*)&lbuf[wave * ND];
  const char* s8 = (const char*)src;
  char* d8 = (char*)dst;
#pragma unroll
  for (int k = 0; k < 3; ++k) {
    __builtin_amdgcn_global_load_async_to_lds_b128(
        (gv4i*)(s8 + lane * 16 + k * 512),
        (lv4i*)(lbase + lane * 16 + k * 512), 0, 0);
  }
  wait_asynccnt0();                         // loads landed in LDS
#pragma unroll
  for (int k = 0; k < 3; ++k) {
    __builtin_amdgcn_global_store_async_from_lds_b128(
        (gv4i*)(d8 + lane * 16 + k * 512),
        (lv4i*)(lbase + lane * 16 + k * 512), 0, 0);
  }
  // S_ENDPGM performs an implicit wait-idle; ASYNCcnt drains before wave exit.
#else
  const float4* s4 = (const float4*)src;
  float4* d4 = (float4*)dst;
  d4[lane]      = s4[lane];
  d4[lane + 32] = s4[lane + 32];
  d4[lane + 64] = s4[lane + 64];
#endif
}

// ---- Host launcher -------------------------------------------------------------
extern "C" void kernel_launch(void* const* d_in, const int* in_sizes, int n_in,
                              void* d_out, int out_size, void* d_ws, size_t ws_size,
                              hipStream_t stream) {
  const float* x   = (const float*)d_in[0];  // (64,512,384) f32
  const int*   dur = (const int*)d_in[1];    // (64,512) i32
  // d_in[2] = max_len scalar (fixed 4096, unused)

  float* out  = (float*)d_out;                       // (64,4096,384) f32
  float* melf = out + (size_t)NB * ML * ND;          // tuple output #2 (64 vals)

  int* csum = (int*)d_ws;                            // NB*NT ints  (128 KB)
  int* mel  = csum + NB * NT;                        // NB ints     (256 B)

  lr_scan<<<NB, NT, 0, stream>>>(dur, csum, mel, melf);
  lr_expand<<<(NB * ML) / 8, 256, 0, stream>>>(x, csum, mel, out);
}